// LossNegSampling_73126113182288
// MI455X (gfx1250) — compile-verified
//
#include <hip/hip_runtime.h>
#include <hip/hip_bf16.h>
#include <math.h>

// ---------------------------------------------------------------------------
// Problem constants (match reference)
// ---------------------------------------------------------------------------
#define B_TOTAL 16384
#define EMB     128
#define NEG     10
#define NLAB    128

typedef __attribute__((ext_vector_type(2))) float v2f;
typedef __attribute__((ext_vector_type(8))) float v8f;

// ws layout (floats): [0] = sum(logsig(pos)+logsig(neg))  [1] = sum(min dist^2)
//                     [2 .. 2+127] = ||emb_com[c]||^2
__device__ __forceinline__ float log_sigmoid(float x) {
    // numerically stable: min(x,0) - log1p(exp(-|x|))
    return fminf(x, 0.0f) - log1pf(expf(-fabsf(x)));
}

// ---------------------------------------------------------------------------
// Kernel 0: zero accumulators + precompute ||com[c]||^2 (64 KB, L2-resident)
// ---------------------------------------------------------------------------
__global__ void ns_init_kernel(const float* __restrict__ com, float* __restrict__ ws) {
    const int t = threadIdx.x;                       // 128 threads
    if (t == 0) { ws[0] = 0.0f; ws[1] = 0.0f; }
    const float* row = com + (size_t)t * EMB;
    float s = 0.0f;
    #pragma unroll 4
    for (int k = 0; k < EMB; ++k) s = fmaf(row[k], row[k], s);
    ws[2 + t] = s;
}

// ---------------------------------------------------------------------------
// Kernel 1: negative-sampling loss. 1 wave32 per sample; 4 f32/lane covers D=128.
// Memory-bound: ~6 KB gathered per sample (96 MB total -> ~4.2 us @ 23.3 TB/s).
// ---------------------------------------------------------------------------
__global__ void ns_negsamp_kernel(const float* __restrict__ emb_u,
                                  const int*   __restrict__ u_node,
                                  const int*   __restrict__ v_node,
                                  const int*   __restrict__ neg_nodes,
                                  float*       __restrict__ ws) {
    const int wave = threadIdx.x >> 5;
    const int lane = threadIdx.x & 31;
    const int b    = blockIdx.x * 8 + wave;          // 8 waves / block

    const float4* urow = (const float4*)(emb_u + (size_t)u_node[b] * EMB);
    const float4* vrow = (const float4*)(emb_u + (size_t)v_node[b] * EMB);
    const float4 uv = urow[lane];
    const float4 vv = vrow[lane];

    // negative_score = sum_n dot(-emb_u[neg_n], u) = -dot(u, sum_n emb_u[neg_n])
    float4 sn = make_float4(0.f, 0.f, 0.f, 0.f);
    #pragma unroll
    for (int n = 0; n < NEG; ++n) {
        const float4* nrow =
            (const float4*)(emb_u + (size_t)neg_nodes[b * NEG + n] * EMB);
        const float4 t = nrow[lane];
        sn.x += t.x; sn.y += t.y; sn.z += t.z; sn.w += t.w;
    }

    float pos = uv.x * vv.x + uv.y * vv.y + uv.z * vv.z + uv.w * vv.w;
    float neg = uv.x * sn.x + uv.y * sn.y + uv.z * sn.z + uv.w * sn.w;
    #pragma unroll
    for (int off = 16; off >= 1; off >>= 1) {
        pos += __shfl_xor(pos, off, 32);
        neg += __shfl_xor(neg, off, 32);
    }
    if (lane == 0) {
        const float ls = log_sigmoid(pos) + log_sigmoid(-neg);
        atomicAdd(&ws[0], ls);
    }
}

// ---------------------------------------------------------------------------
// Kernel 2: clustering. 256 threads (8 waves) per block, 16 samples per block.
// dist^2 = ||u||^2 + ||c||^2 - 2 * (U x Com^T) ; the GEMM runs on
// V_WMMA_F32_16X16X4_F32 (exact f32 matrix path). Each wave owns a 16x16
// output tile (wave w -> columns w*16..w*16+15), K=128 in 32 WMMA steps.
// ---------------------------------------------------------------------------
__global__ void ns_cluster_kernel(const float* __restrict__ emb_u,
                                  const float* __restrict__ com,
                                  const int*   __restrict__ u_node,
                                  const float* __restrict__ com_norm,  // ws+2
                                  float*       __restrict__ ws,
                                  float*       __restrict__ out) {
    // Row pad 128 -> 132 floats (132 % 64 banks == 4) => conflict-free
    // column-ish A-fragment reads and disjoint-bank tile writes.
    __shared__ __align__(16) float u_tile[16][132];
    __shared__ __align__(16) float d2s[16][132];
    __shared__ float normU[16];

    const int tid    = threadIdx.x;
    const int wave   = tid >> 5;
    const int lane   = tid & 31;
    const int b_base = blockIdx.x * 16;

    // --- Phase 1: gather 16 U rows into LDS (2048 floats, 8 per thread) ---
    {
        const int r = tid >> 4;          // 0..15  (row)
        const int c = (tid & 15) * 8;    // 0..120 (col, 8-float segment)
        const float* src = emb_u + (size_t)u_node[b_base + r] * EMB + c;
        const float4 a0 = ((const float4*)src)[0];
        const float4 a1 = ((const float4*)src)[1];
        *(float4*)&u_tile[r][c + 0] = a0;    // 528-byte row stride: 16B aligned
        *(float4*)&u_tile[r][c + 4] = a1;
    }
    __syncthreads();

    if (tid < 16) {
        float s = 0.0f;
        #pragma unroll 4
        for (int k = 0; k < EMB; ++k) { const float x = u_tile[tid][k]; s = fmaf(x, x, s); }
        normU[tid] = s;
    }
    __syncthreads();

    // --- Phase 2: WMMA f32 16x16x4, K = 128 ---
    // A 16x4 layout: lanes 0-15 hold {K0,K1} of row m=lane, lanes 16-31 hold {K2,K3}.
    // B 4x16 layout mirrors A with n=lane&15; B[k][n] = Com[n][k] (dot of rows).
    const int m     = lane & 15;
    const int khalf = lane >> 4;               // 0 or 1
    const int n     = wave * 16 + m;           // output column of this lane
    const float* brow = com + (size_t)n * EMB;

    v8f acc = {};
    #pragma unroll
    for (int k0 = 0; k0 < EMB; k0 += 4) {
        const int k = k0 + 2 * khalf;
        v2f a, b;
        a.x = u_tile[m][k];     a.y = u_tile[m][k + 1];   // LDS, conflict-free
        b.x = brow[k];          b.y = brow[k + 1];        // L2-resident (64 KB)
        acc = __builtin_amdgcn_wmma_f32_16x16x4_f32(
            /*neg_a=*/false, a, /*neg_b=*/false, b,
            /*c_mod=*/(short)0, acc, /*reuse_a=*/false, /*reuse_b=*/false);
    }

    // --- Phase 3: dist^2 into LDS tile ---
    // D layout: lanes 0-15 -> M = j, lanes 16-31 -> M = j + 8; N = lane&15 (+wave*16).
    const float nC = com_norm[n];
    #pragma unroll
    for (int j = 0; j < 8; ++j) {
        const int row = j + 8 * khalf;
        const float d2 = fmaxf(normU[row] + nC - 2.0f * acc[j], 0.0f);
        d2s[row][wave * 16 + m] = d2;
    }
    __syncthreads();

    // --- Phase 4: per-row argmin over 128 labels ---
    if (tid < 16) {
        float best = d2s[tid][0];
        int   bi   = 0;
        #pragma unroll 4
        for (int c = 1; c < NLAB; ++c) {
            const float v = d2s[tid][c];
            if (v < best) { best = v; bi = c; }   // first-min tie-break (argmin)
        }
        out[1 + b_base + tid] = (float)bi;         // cluster_choice
        atomicAdd(&ws[1], best);                   // min(dist)^2 == min(dist^2)
    }
}

// ---------------------------------------------------------------------------
// Kernel 3: final_loss = -mean(logsig sums) + mean(min dist^2)
// ---------------------------------------------------------------------------
__global__ void ns_finalize_kernel(const float* __restrict__ ws,
                                   float* __restrict__ out) {
    const float invB = 1.0f / (float)B_TOTAL;
    out[0] = -(ws[0] * invB) + ws[1] * invB;
}

// ---------------------------------------------------------------------------
extern "C" void kernel_launch(void* const* d_in, const int* in_sizes, int n_in,
                              void* d_out, int out_size, void* d_ws, size_t ws_size,
                              hipStream_t stream) {
    const float* emb_u     = (const float*)d_in[0];   // (100000,128) f32
    const float* emb_com   = (const float*)d_in[1];   // (128,128)    f32
    const int*   u_node    = (const int*)  d_in[2];   // (16384,1)
    const int*   v_node    = (const int*)  d_in[3];   // (16384,1)
    const int*   neg_nodes = (const int*)  d_in[4];   // (16384,10)
    float* ws  = (float*)d_ws;
    float* out = (float*)d_out;

    ns_init_kernel   <<<1,            128, 0, stream>>>(emb_com, ws);
    ns_negsamp_kernel<<<B_TOTAL / 8,  256, 0, stream>>>(emb_u, u_node, v_node,
                                                        neg_nodes, ws);
    ns_cluster_kernel<<<B_TOTAL / 16, 256, 0, stream>>>(emb_u, emb_com, u_node,
                                                        ws + 2, ws, out);
    ns_finalize_kernel<<<1, 1, 0, stream>>>(ws, out);
}